// IRLLinearModel_42786464203479
// MI455X (gfx1250) — compile-verified
//
#include <hip/hip_runtime.h>

typedef __attribute__((ext_vector_type(2))) float v2f;
typedef __attribute__((ext_vector_type(4))) float v4f;
typedef __attribute__((ext_vector_type(8))) float v8f;

#define Bn 64
#define Ln 200
#define Fn 64
#define LDS_STRIDE 68   // 64 + 4 pad -> conflict-free 16-row b64 reads

// One workgroup (128 threads / 4 waves) handles one (i,j) position:
//   out[0:64, i, j] = input[0:64, i, j, :] (64x64)  .  theta[i, j, :] (64)
// lifted to WMMA: C(16x16) = A(16x64) x B(64x16) with theta replicated
// across all 16 N-columns; every C column holds the answer, lane N=0 stores.
__global__ __launch_bounds__(128) void irl_pos_dot_wmma(
    const float* __restrict__ inp,    // [B, L, L, F]
    const float* __restrict__ theta,  // [L, L, F]
    float* __restrict__ out)          // [B, L, L]
{
    __shared__ float sA[Bn * LDS_STRIDE];  // 64 batch rows x 64 f (padded)
    __shared__ float sT[Fn];               // theta for this (i,j)

    const int pos = blockIdx.x;            // i*L + j  (0..39999)
    const int tid = threadIdx.x;           // 0..127

    const size_t posBase     = (size_t)pos * Fn;           // offset of (i,j,0)
    const size_t batchStride = (size_t)Ln * Ln * Fn;       // elements per batch

    // ---- Stage input tile: 128 threads, 16 threads per batch row (b128 each),
    //      8 rows per pass, 8 passes -> 64 rows. 256B contiguous per row.
    {
        const int lrow = tid >> 4;            // 0..7
        const int lcol = (tid & 15) << 2;     // 0,4,...,60
        #pragma unroll
        for (int p = 0; p < 8; ++p) {
            const int b = p * 8 + lrow;
            v4f v = *(const v4f*)(inp + (size_t)b * batchStride + posBase + lcol);
            *(v4f*)(sA + b * LDS_STRIDE + lcol) = v;
        }
        if (tid < 16) {
            v4f t = *(const v4f*)(theta + posBase + (tid << 2));
            *(v4f*)(sT + (tid << 2)) = t;
        }
    }
    __syncthreads();

    // ---- Compute: wave w handles batches [16w, 16w+16).
    const int wave  = tid >> 5;            // 0..3
    const int lane  = tid & 31;
    const int m     = lane & 15;           // A row within tile (C column)
    const int khalf = (lane >> 4) << 1;    // lanes 0-15 -> K+{0,1}, 16-31 -> K+{2,3}

    const float* aRow = sA + (wave * 16 + m) * LDS_STRIDE;

    v8f c = {};
    #pragma unroll
    for (int kk = 0; kk < 16; ++kk) {
        const int k = (kk << 2) + khalf;
        // A fragment (ISA layout): lane m VGPR0/1 = A[m][k], A[m][k+1]
        v2f a = *(const v2f*)(aRow + k);
        // B fragment: theta replicated over N -> every lane needs theta[k],theta[k+1]
        // (half-wave broadcast from LDS, conflict-free)
        v2f b = *(const v2f*)(sT + k);
        // D = A*B + C  : v_wmma_f32_16x16x4_f32
        c = __builtin_amdgcn_wmma_f32_16x16x4_f32(
                /*neg_a=*/false, a, /*neg_b=*/false, b,
                /*c_mod=*/(short)0, c, /*reuse_a=*/false, /*reuse_b=*/false);
    }

    // ---- Store: all C columns identical; take N=0 (lane 0 -> M=0..7,
    //      lane 16 -> M=8..15 per 32-bit C/D layout).
    if ((lane & 15) == 0) {
        const int b0 = wave * 16 + ((lane >> 4) << 3);
        #pragma unroll
        for (int r = 0; r < 8; ++r) {
            out[(size_t)(b0 + r) * (Ln * Ln) + pos] = c[r];
        }
    }
}

extern "C" void kernel_launch(void* const* d_in, const int* in_sizes, int n_in,
                              void* d_out, int out_size, void* d_ws, size_t ws_size,
                              hipStream_t stream) {
    const float* inp   = (const float*)d_in[0];   // [64,200,200,64] f32
    const float* theta = (const float*)d_in[1];   // [200,200,64]    f32
    float*       out   = (float*)d_out;           // [64,200,200]    f32

    dim3 grid(Ln * Ln);   // 40000 positions
    dim3 block(128);      // 4 wave32
    irl_pos_dot_wmma<<<grid, block, 0, stream>>>(inp, theta, out);
}